// NeoGNN_9887014715909
// MI455X (gfx1250) — compile-verified
//
#include <hip/hip_runtime.h>
#include <hip/hip_bf16.h>

typedef __attribute__((ext_vector_type(2))) float v2f;
typedef __attribute__((ext_vector_type(8))) float v8f;

#define F_DIM 256
#define D_DEG 32

// ---------------------------------------------------------------------------
// Stage 1: nsf[c] = sum over all nnz with col==c of f_edge(nbr_w)  (scatter)
// f_edge: Linear(1,8) -> ReLU -> Linear(8,1)
// ---------------------------------------------------------------------------
__global__ __launch_bounds__(256) void neognn_scatter_fe(
    const float* __restrict__ nbr_w, const int* __restrict__ nbr_idx,
    const float* __restrict__ fe_w1, const float* __restrict__ fe_b1,
    const float* __restrict__ fe_w2, const float* __restrict__ fe_b2,
    float* __restrict__ nsf, int total)
{
    int i = blockIdx.x * blockDim.x + threadIdx.x;
    if (i >= total) return;
    float x = nbr_w[i];
    float acc = fe_b2[0];
#pragma unroll
    for (int j = 0; j < 8; ++j)
        acc = fmaf(fe_w2[j], fmaxf(fmaf(x, fe_w1[j], fe_b1[j]), 0.0f), acc);
    atomicAdd(&nsf[nbr_idx[i]], acc);
}

// ---------------------------------------------------------------------------
// Stage 2: edge-0 struct score. One wave (32 lanes == D).
//   lane l: handles column l of src row and dst row of edge 0.
//   fnode(c) = f_node MLP (hidden 128) evaluated lazily at needed cols only.
//   sorted-intersection dot == searchsorted(side='left') + clamp + match.
//   lane 0: s0 = sigmoid(g_phi(dot)); a = softmax(alpha);
//           scalars[0] = a0*s0 + 1e-15 ; scalars[1] = a1
// ---------------------------------------------------------------------------
__global__ __launch_bounds__(32) void neognn_struct_edge0(
    const int* __restrict__ edge, const int* __restrict__ nbr_idx,
    const float* __restrict__ nbr_w, const float* __restrict__ nsf,
    const float* __restrict__ fn_w1, const float* __restrict__ fn_b1,
    const float* __restrict__ fn_w2, const float* __restrict__ fn_b2,
    const float* __restrict__ gp_w1, const float* __restrict__ gp_b1,
    const float* __restrict__ gp_w2, const float* __restrict__ gp_b2,
    const float* __restrict__ alpha, float* __restrict__ scalars, int B)
{
    __shared__ int   s_cols[D_DEG];
    __shared__ float s_w[D_DEG];
    __shared__ float s_part[D_DEG];

    const int lane = threadIdx.x;
    const int r0 = edge[0];
    const int r1 = edge[B];

    const int   cs  = nbr_idx[r0 * D_DEG + lane];
    const float wsv = nbr_w[r0 * D_DEG + lane];
    const int   cd  = nbr_idx[r1 * D_DEG + lane];
    const float wdv = nbr_w[r1 * D_DEG + lane];

    // f_node MLP at a column (input = nsf[col], hidden 128)
    auto fnode = [&](int col) {
        float x = nsf[col];
        float acc = fn_b2[0];
#pragma unroll 8
        for (int j = 0; j < 128; ++j)
            acc = fmaf(fn_w2[j], fmaxf(fmaf(x, fn_w1[j], fn_b1[j]), 0.0f), acc);
        return acc;
    };

    const float w_src = wsv * fnode(cs);
    const float w_dst = wdv * fnode(cd);

    s_cols[lane] = cd;
    s_w[lane]    = w_dst;
    __syncthreads();

    // lower_bound (== jnp.searchsorted side='left')
    int lo = 0, hi = D_DEG;
    while (lo < hi) {
        int mid = (lo + hi) >> 1;
        if (s_cols[mid] < cs) lo = mid + 1; else hi = mid;
    }
    int pos = min(lo, D_DEG - 1);
    float contrib = (s_cols[pos] == cs) ? (w_src * s_w[pos]) : 0.0f;
    s_part[lane] = contrib;
    __syncthreads();

    if (lane == 0) {
        float dot = 0.0f;
#pragma unroll
        for (int i = 0; i < D_DEG; ++i) dot += s_part[i];
        float acc = gp_b2[0];
#pragma unroll 8
        for (int j = 0; j < 128; ++j)
            acc = fmaf(gp_w2[j], fmaxf(fmaf(dot, gp_w1[j], gp_b1[j]), 0.0f), acc);
        float s0 = 1.0f / (1.0f + __expf(-acc));
        float m  = fmaxf(alpha[0], alpha[1]);
        float e0 = __expf(alpha[0] - m), e1 = __expf(alpha[1] - m);
        float inv = 1.0f / (e0 + e1);
        scalars[0] = (e0 * inv) * s0 + 1e-15f;   // c0: broadcast struct term
        scalars[1] = (e1 * inv);                 // a1: feature-term weight
    }
}

// ---------------------------------------------------------------------------
// Stage 3: out[b] = a1 * dot(z[e0[b]], z[e1[b]]) + c0
// One wave per 16 edges: 16x16 cross-dot block via V_WMMA_F32_16X16X4_F32,
// K = 256 in 64 steps; take the diagonal.
//   A (16x4 f32): lane l -> row = l&15, holds K = 2*(l>>4) + {0,1}
//   B (4x16 f32): lane l -> col = l&15, holds K = 2*(l>>4) + {0,1}
// So both operands use identical per-lane addressing on src/dst rows.
// Diagonal (i,i): i<8 -> lane i, c[i];  i>=8 -> lane i+16, c[i-8].
// ---------------------------------------------------------------------------
__global__ __launch_bounds__(256) void neognn_feat_wmma(
    const float* __restrict__ z, const int* __restrict__ edge,
    const float* __restrict__ scalars, float* __restrict__ out, int B)
{
    const int lane  = threadIdx.x & 31;
    const int wave  = threadIdx.x >> 5;
    const int group = blockIdx.x * 8 + wave;     // 16 edges per wave
    const int base  = group * 16;
    if (base >= B) return;                       // wave-uniform: EXEC stays full

    const int row   = lane & 15;
    const int khalf = lane >> 4;

    const int r0 = edge[base + row];
    const int r1 = edge[B + base + row];

    const float* pa = z + (size_t)r0 * F_DIM + 2 * khalf;
    const float* pb = z + (size_t)r1 * F_DIM + 2 * khalf;

    v8f c = {0.f, 0.f, 0.f, 0.f, 0.f, 0.f, 0.f, 0.f};
#pragma unroll 4
    for (int kk = 0; kk < F_DIM / 4; ++kk) {
        v2f a = *(const v2f*)(pa + 4 * kk);
        v2f b = *(const v2f*)(pb + 4 * kk);
        // 8 args: (neg_a, A, neg_b, B, c_mod, C, reuse_a, reuse_b)
        c = __builtin_amdgcn_wmma_f32_16x16x4_f32(
                false, a, false, b, (short)0, c, false, false);
    }

    float diag = 0.0f;
#pragma unroll
    for (int r = 0; r < 8; ++r)
        diag = (lane == r || lane == 24 + r) ? c[r] : diag;

    const float c0 = scalars[0];
    const float a1 = scalars[1];
    if (lane < 8)
        out[base + lane] = fmaf(a1, diag, c0);
    else if (lane >= 24)
        out[base + lane - 16] = fmaf(a1, diag, c0);
}

// ---------------------------------------------------------------------------
extern "C" void kernel_launch(void* const* d_in, const int* in_sizes, int n_in,
                              void* d_out, int out_size, void* d_ws, size_t ws_size,
                              hipStream_t stream) {
    const float* z       = (const float*)d_in[0];
    const int*   edge    = (const int*)d_in[1];
    const int*   nbr_idx = (const int*)d_in[2];
    const float* nbr_w   = (const float*)d_in[3];
    const float* fe_w1   = (const float*)d_in[4];
    const float* fe_b1   = (const float*)d_in[5];
    const float* fe_w2   = (const float*)d_in[6];
    const float* fe_b2   = (const float*)d_in[7];
    const float* fn_w1   = (const float*)d_in[8];
    const float* fn_b1   = (const float*)d_in[9];
    const float* fn_w2   = (const float*)d_in[10];
    const float* fn_b2   = (const float*)d_in[11];
    const float* gp_w1   = (const float*)d_in[12];
    const float* gp_b1   = (const float*)d_in[13];
    const float* gp_w2   = (const float*)d_in[14];
    const float* gp_b2   = (const float*)d_in[15];
    const float* alpha   = (const float*)d_in[16];

    const int N = in_sizes[0] / F_DIM;     // 50000
    const int B = in_sizes[1] / 2;         // 65536
    const int total = N * D_DEG;           // 1.6M nnz

    float* scalars = (float*)d_ws;                       // [0]=c0, [1]=a1
    float* nsf     = (float*)((char*)d_ws + 16);         // N floats

    hipMemsetAsync(nsf, 0, (size_t)N * sizeof(float), stream);

    neognn_scatter_fe<<<(total + 255) / 256, 256, 0, stream>>>(
        nbr_w, nbr_idx, fe_w1, fe_b1, fe_w2, fe_b2, nsf, total);

    neognn_struct_edge0<<<1, 32, 0, stream>>>(
        edge, nbr_idx, nbr_w, nsf,
        fn_w1, fn_b1, fn_w2, fn_b2,
        gp_w1, gp_b1, gp_w2, gp_b2,
        alpha, scalars, B);

    const int groups = B / 16;                           // 4096 waves
    neognn_feat_wmma<<<(groups + 7) / 8, 256, 0, stream>>>(
        z, edge, scalars, (float*)d_out, B);
}